// DsbnKdLoss_16638703305205
// MI455X (gfx1250) — compile-verified
//
#include <hip/hip_runtime.h>
#include <hip/hip_bf16.h>
#include <math.h>

typedef float v2f __attribute__((ext_vector_type(2)));
typedef float v8f __attribute__((ext_vector_type(8)));

#define HWZ   2457600     // 160*160*96
#define QHWZ  614400      // HWZ/4
#define NGRP  2457600     // (4*HWZ)/4 groups of 4 voxels
#define NB    1024
#define NT    256
#define KD_EPS 1e-6f

// ---------------- init: zero the 40-float accumulator ----------------
__global__ void kd_zero40(float* __restrict__ acc) {
  if (threadIdx.x < 40) acc[threadIdx.x] = 0.f;
}

// ---------------- main streaming reduction (memory-bound) ----------------
// acc layout per side (20 floats): [0..15] sums[i][k] (i=mask class, k=channel), [16..19] counts[i]
__global__ __launch_bounds__(NT) void kd_class_sums(
    const float* __restrict__ Ls, const float* __restrict__ Lt,
    const int*   __restrict__ Gs, const int*   __restrict__ Gt,
    float* __restrict__ acc)
{
  const int side = blockIdx.y;
  const float* __restrict__ L = side ? Lt : Ls;
  const int*   __restrict__ G = side ? Gt : Gs;

  float s[4][4];
  float cnt[4];
#pragma unroll
  for (int i = 0; i < 4; ++i) {
    cnt[i] = 0.f;
#pragma unroll
    for (int k = 0; k < 4; ++k) s[i][k] = 0.f;
  }

  const int tid    = threadIdx.x;
  const int stride = NB * NT;

  for (int g = blockIdx.x * NT + tid; g < NGRP; g += stride) {
    const int b    = g / QHWZ;             // batch index
    const int r    = (g - b * QHWZ) * 4;   // offset within (H,W,Z) block
    const int base = b * (4 * HWZ) + r;    // channel-0 element offset (fits in int)

    const float4 x0 = *(const float4*)(L + base);
    const float4 x1 = *(const float4*)(L + base + HWZ);
    const float4 x2 = *(const float4*)(L + base + 2 * HWZ);
    const float4 x3 = *(const float4*)(L + base + 3 * HWZ);
    const int4   lb = *(const int4*)(G + 4 * g);

    // speculative prefetch of next grid-stride iteration (global_prefetch_b8)
    {
      const int gn = min(g + stride, NGRP - 1);
      const int bn = gn / QHWZ;
      const int pb = bn * (4 * HWZ) + (gn - bn * QHWZ) * 4;
      __builtin_prefetch(L + pb,          0, 0);
      __builtin_prefetch(L + pb + HWZ,    0, 0);
      __builtin_prefetch(L + pb + 2*HWZ,  0, 0);
      __builtin_prefetch(L + pb + 3*HWZ,  0, 0);
      __builtin_prefetch(G + 4 * gn,      0, 0);
    }

    const int   labs[4]   = { lb.x, lb.y, lb.z, lb.w };
    const float vox[4][4] = { { x0.x, x1.x, x2.x, x3.x },
                              { x0.y, x1.y, x2.y, x3.y },
                              { x0.z, x1.z, x2.z, x3.z },
                              { x0.w, x1.w, x2.w, x3.w } };
#pragma unroll
    for (int j = 0; j < 4; ++j) {
      const int lab = labs[j];
#pragma unroll
      for (int i = 0; i < 4; ++i) {
        const float f = (lab == i) ? 1.f : 0.f;   // v_cmp + v_cndmask
        cnt[i] += f;
#pragma unroll
        for (int k = 0; k < 4; ++k) s[i][k] = fmaf(f, vox[j][k], s[i][k]);
      }
    }
  }

  // ---- block reduction in LDS ----
  __shared__ float sm[NT * 20];
#pragma unroll
  for (int i = 0; i < 4; ++i) {
#pragma unroll
    for (int k = 0; k < 4; ++k) sm[tid * 20 + i * 4 + k] = s[i][k];
    sm[tid * 20 + 16 + i] = cnt[i];
  }
  __syncthreads();
  for (int st = NT / 2; st > 0; st >>= 1) {
    if (tid < st) {
#pragma unroll
      for (int v = 0; v < 20; ++v) sm[tid * 20 + v] += sm[(tid + st) * 20 + v];
    }
    __syncthreads();
  }
  if (tid < 20) atomicAdd(acc + side * 20 + tid, sm[tid]);
}

// ---------------- epilogue: WMMA row-scaling + softmax + symmetric KL ----------------
// acc: [0..19] src (sums,counts), [20..39] trg (sums,counts). One wave.
__global__ __launch_bounds__(32) void kd_final(const float* __restrict__ acc,
                                               float* __restrict__ out)
{
  __shared__ float A[40];
  __shared__ float Dsh[16][16];

  const int t = threadIdx.x;                // 0..31
  A[t] = acc[t];
  const int u = (t + 32 < 40) ? (t + 32) : 0;   // clamped, race-safe (same value)
  A[u] = acc[u];
  __syncthreads();

  // per-lane reciprocal counts (all lanes, no divergence)
  float rs[4], rt[4];
#pragma unroll
  for (int i = 0; i < 4; ++i) {
    rs[i] = 1.f / (A[16 + i] + KD_EPS);
    rt[i] = 1.f / (A[36 + i] + KD_EPS);
  }

  const int hi = t >> 4;      // 0: lanes 0-15 (K=0/1), 1: lanes 16-31 (K=2/3)
  const int m  = t & 15;      // A-row / B-col index held by this lane
  const int k0 = hi ? 2 : 0;
  const int k1 = hi ? 3 : 1;

  // A (16x4): rows 0-3 diag(recip_src), rows 4-7 diag(recip_trg)
  const float rs_k0 = hi ? rs[2] : rs[0];
  const float rt_k0 = hi ? rt[2] : rt[0];
  const float rs_k1 = hi ? rs[3] : rs[1];
  const float rt_k1 = hi ? rt[3] : rt[1];
  const float a0 = (m == k0) ? rs_k0 : ((m == k0 + 4) ? rt_k0 : 0.f);
  const float a1 = (m == k1) ? rs_k1 : ((m == k1 + 4) ? rt_k1 : 0.f);

  // B (4x16): row k, cols 0-3 = sums_src[k][*], cols 4-7 = sums_trg[k][*]
  const float b0 = (m < 4) ? A[k0 * 4 + m] : ((m < 8) ? A[20 + k0 * 4 + (m - 4)] : 0.f);
  const float b1 = (m < 4) ? A[k1 * 4 + m] : ((m < 8) ? A[20 + k1 * 4 + (m - 4)] : 0.f);

  v2f av; av[0] = a0; av[1] = a1;
  v2f bv; bv[0] = b0; bv[1] = b1;
  v8f cv = {};
  // D = A x B : D[0:4,0:4] = avg_src, D[4:8,4:8] = avg_trg
  cv = __builtin_amdgcn_wmma_f32_16x16x4_f32(
      /*neg_a=*/false, av, /*neg_b=*/false, bv,
      /*c_mod=*/(short)0, cv, /*reuse_a=*/false, /*reuse_b=*/false);

  // spill D (M = hi*8 + p, N = m)
#pragma unroll
  for (int p = 0; p < 8; ++p) Dsh[hi * 8 + p][m] = cv[p];
  __syncthreads();

  if (t == 0) {
    float sp[4][4], tp[4][4];
#pragma unroll
    for (int i = 0; i < 4; ++i) {
      float vs[4], vt[4];
#pragma unroll
      for (int j = 0; j < 4; ++j) {
        vs[j] = Dsh[i][j] * 0.5f;           // avg_src / TEMPERATURE
        vt[j] = Dsh[4 + i][4 + j] * 0.5f;   // avg_trg / TEMPERATURE
      }
      float ms = vs[0], mt = vt[0];
#pragma unroll
      for (int j = 1; j < 4; ++j) { ms = fmaxf(ms, vs[j]); mt = fmaxf(mt, vt[j]); }
      float Ss = 0.f, St = 0.f;
#pragma unroll
      for (int j = 0; j < 4; ++j) {
        sp[i][j] = expf(vs[j] - ms); Ss += sp[i][j];
        tp[i][j] = expf(vt[j] - mt); St += tp[i][j];
      }
      const float iSs = 1.f / Ss, iSt = 1.f / St;
#pragma unroll
      for (int j = 0; j < 4; ++j) { sp[i][j] *= iSs; tp[i][j] *= iSt; }
    }
    float kl = 0.f;
#pragma unroll
    for (int i = 0; i < 4; ++i)
#pragma unroll
      for (int j = 0; j < 4; ++j) {
        const float r = logf(sp[i][j] / tp[i][j]);
        kl += sp[i][j] * r - tp[i][j] * r;   // sp*log(sp/tp) + tp*log(tp/sp)
      }
    out[0] = kl * 0.125f;                    // /2 /C (C=4)
  }
}

extern "C" void kernel_launch(void* const* d_in, const int* in_sizes, int n_in,
                              void* d_out, int out_size, void* d_ws, size_t ws_size,
                              hipStream_t stream) {
  const float* Ls = (const float*)d_in[0];
  const float* Lt = (const float*)d_in[1];
  const int*   Gs = (const int*)d_in[2];
  const int*   Gt = (const int*)d_in[3];
  float* out = (float*)d_out;
  float* acc = (float*)d_ws;   // 40 floats

  kd_zero40<<<1, 64, 0, stream>>>(acc);
  dim3 grid(NB, 2, 1);         // grid.y: 0 = src, 1 = trg
  kd_class_sums<<<grid, NT, 0, stream>>>(Ls, Lt, Gs, Gt, acc);
  kd_final<<<1, 32, 0, stream>>>(acc, out);
}